// GGNNModel_85770496901353
// MI455X (gfx1250) — compile-verified
//
#include <hip/hip_runtime.h>
#include <math.h>

#define N_NODES 100000
#define N_EDGES 1600000
#define HIDDEN 128
#define NUM_CLASSES 10
#define NUM_LAYERS 4
#define NUM_GRAPHS 200

typedef float v2f __attribute__((ext_vector_type(2)));
typedef float v8f __attribute__((ext_vector_type(8)));

// D = A(16x4) * B(4x16) + C(16x16), exact fp32 via CDNA5 WMMA.
// args: (neg_a, A, neg_b, B, c_mod, C, reuse_a, reuse_b)
__device__ __forceinline__ v8f wmma_f32_k4(v2f a, v2f b, v8f c) {
    return __builtin_amdgcn_wmma_f32_16x16x4_f32(false, a, false, b, (short)0, c,
                                                 false, false);
}

#define LDS_STRIDE 132  // 128 + 4 pad -> bank-conflict-free M-striped reads

// ---------------------------------------------------------------------------
// Packed-B layout: for pair-row kp (= k/2) and column n (of P columns):
//   Bp[kp*(2P) + 2n + 0] = B[2kp  ][n]
//   Bp[kp*(2P) + 2n + 1] = B[2kp+1][n]
// -> one aligned 8-byte load per lane yields the v2f WMMA B operand
//    (contiguous 128B per 16-lane half-wave).
// ---------------------------------------------------------------------------

// Pack from "direct" layout: src is [K=128][P] row-major (B[k][n] = src[k*P+n])
__global__ void k_pack_direct(const float* __restrict__ src,
                              float* __restrict__ dst, int P) {
    const int i = blockIdx.x * 256 + threadIdx.x;
    if (i >= 128 * P) return;
    const int k = i / P, n = i % P;
    dst[(k >> 1) * (2 * P) + 2 * n + (k & 1)] = src[i];
}

// Pack from torch layout: src is [P][K=128] row-major (B[k][n] = src[n*128+k])
__global__ void k_pack_torch(const float* __restrict__ src,
                             float* __restrict__ dst, int P) {
    const int i = blockIdx.x * 256 + threadIdx.x;
    if (i >= P * 128) return;
    const int n = i / 128, k = i % 128;
    dst[(k >> 1) * (2 * P) + 2 * n + (k & 1)] = src[i];
}

// ---------------------------------------------------------------------------
// C[N,128] = A[N,128] @ B(packed, P=128 cols) (+bias)
// grid.x = N/16 row-blocks, 256 threads = 8 waves, wave w owns cols [16w,16w+16)
// ---------------------------------------------------------------------------
__global__ __launch_bounds__(256) void k_gemm_nx128(
    const float* __restrict__ A, const float* __restrict__ Bp,
    const float* __restrict__ bias, float* __restrict__ C) {
    __shared__ float As[16 * LDS_STRIDE];
    const int tid = threadIdx.x;
    const int rowbase = blockIdx.x * 16;

    for (int i = tid; i < 16 * 128; i += 256)
        As[(i >> 7) * LDS_STRIDE + (i & 127)] = A[rowbase * 128 + i];
    __syncthreads();

    const int lane = tid & 31;
    const int wave = tid >> 5;
    const int c0 = wave * 16;
    const int r = lane & 15;             // A row
    const int n = lane & 15;             // B col within tile
    const int khalf = (lane >> 4) << 1;  // 0 or 2

    // per-lane packed-B pointer: pair-row (khalf>>1), column c0+n
    const float* pB = Bp + (khalf >> 1) * 256 + (c0 + n) * 2;

    v8f acc = {0.f, 0.f, 0.f, 0.f, 0.f, 0.f, 0.f, 0.f};
    for (int kb = 0; kb < 128; kb += 4) {
        const int k0 = kb + khalf;
        v2f a;
        a.x = As[r * LDS_STRIDE + k0];
        a.y = As[r * LDS_STRIDE + k0 + 1];
        const v2f b = *(const v2f*)pB;
        pB += 4 * 128;  // kb+=4 -> pair-row += 2 -> 2*(2P) floats
        acc = wmma_f32_k4(a, b, acc);
    }

    const int rh = (lane >> 4) * 8;
    const float bval = bias ? bias[c0 + n] : 0.f;
    for (int j = 0; j < 8; ++j)
        C[(rowbase + rh + j) * 128 + c0 + n] = acc[j] + bval;
}

// ---------------------------------------------------------------------------
// Fused GRU: h = GRUCell(agg, h). Six WMMA accumulator chains per wave
// (i_r,i_z,i_n,h_r,h_z,h_n) -> gate matrices never hit memory.
// Weights pre-packed with P=384 (r-gate cols 0..127, z 128..255, n 256..383).
// ---------------------------------------------------------------------------
__global__ __launch_bounds__(256) void k_gru_fused(
    const float* __restrict__ agg, float* __restrict__ h,
    const float* __restrict__ wihP, const float* __restrict__ whhP,
    const float* __restrict__ bih, const float* __restrict__ bhh) {
    __shared__ float Aagg[16 * LDS_STRIDE];
    __shared__ float Ah[16 * LDS_STRIDE];
    const int tid = threadIdx.x;
    const int rowbase = blockIdx.x * 16;

    for (int i = tid; i < 16 * 128; i += 256) {
        const int ro = (i >> 7) * LDS_STRIDE + (i & 127);
        Aagg[ro] = agg[rowbase * 128 + i];
        Ah[ro] = h[rowbase * 128 + i];
    }
    __syncthreads();

    const int lane = tid & 31;
    const int wave = tid >> 5;
    const int c0 = wave * 16;
    const int r = lane & 15;
    const int n = lane & 15;
    const int khalf = (lane >> 4) << 1;

    // packed pitch 2*384 = 768 floats per pair-row; gate offsets in floats:
    // r-gate at +0, z at +256 (=128 cols*2), n at +512
    const float* pI = wihP + (khalf >> 1) * 768 + (c0 + n) * 2;
    const float* pH = whhP + (khalf >> 1) * 768 + (c0 + n) * 2;

    v8f cir = {0.f, 0.f, 0.f, 0.f, 0.f, 0.f, 0.f, 0.f};
    v8f ciz = cir, cin = cir, chr_ = cir, chz = cir, chn = cir;

    for (int kb = 0; kb < 128; kb += 4) {
        const int k0 = kb + khalf;
        v2f aa, ah;
        aa.x = Aagg[r * LDS_STRIDE + k0];
        aa.y = Aagg[r * LDS_STRIDE + k0 + 1];
        ah.x = Ah[r * LDS_STRIDE + k0];
        ah.y = Ah[r * LDS_STRIDE + k0 + 1];

        const v2f bir = *(const v2f*)(pI);
        const v2f biz = *(const v2f*)(pI + 256);
        const v2f bin = *(const v2f*)(pI + 512);
        const v2f bhr = *(const v2f*)(pH);
        const v2f bhz = *(const v2f*)(pH + 256);
        const v2f bhn = *(const v2f*)(pH + 512);
        pI += 4 * 384;  // pair-row += 2
        pH += 4 * 384;

        cir = wmma_f32_k4(aa, bir, cir);
        ciz = wmma_f32_k4(aa, biz, ciz);
        cin = wmma_f32_k4(aa, bin, cin);
        chr_ = wmma_f32_k4(ah, bhr, chr_);
        chz = wmma_f32_k4(ah, bhz, chz);
        chn = wmma_f32_k4(ah, bhn, chn);
    }

    const int rh = (lane >> 4) * 8;
    const int c = c0 + n;
    const float b_ir = bih[c], b_hr = bhh[c];
    const float b_iz = bih[c + 128], b_hz = bhh[c + 128];
    const float b_in = bih[c + 256], b_hn = bhh[c + 256];
    for (int j = 0; j < 8; ++j) {
        const int m = rh + j;
        const float rg = 1.f / (1.f + expf(-(cir[j] + b_ir + chr_[j] + b_hr)));
        const float zg = 1.f / (1.f + expf(-(ciz[j] + b_iz + chz[j] + b_hz)));
        const float ng = tanhf(cin[j] + b_in + rg * (chn[j] + b_hn));
        const float ho = Ah[m * LDS_STRIDE + c];
        h[(rowbase + m) * 128 + c] = (1.f - zg) * ng + zg * ho;
    }
}

// ---------------------------------------------------------------------------
// Edge scatter-add: one wave per edge, 4 floats/lane; atomics land in L2
// (agg = 51 MB, fits the 192 MB L2).
// ---------------------------------------------------------------------------
__global__ __launch_bounds__(256) void k_scatter(
    const float* __restrict__ m, const int* __restrict__ ei,
    float* __restrict__ agg, int nEdges) {
    const int gid = blockIdx.x * 256 + threadIdx.x;
    const int e = gid >> 5;
    if (e >= nEdges) return;
    const int lane = gid & 31;
    const int src = ei[e];
    const int dst = ei[nEdges + e];
    const float4 mv = *(const float4*)(m + src * 128 + lane * 4);
    float* p = agg + dst * 128 + lane * 4;
    atomicAdd(p + 0, mv.x);
    atomicAdd(p + 1, mv.y);
    atomicAdd(p + 2, mv.z);
    atomicAdd(p + 3, mv.w);
}

__global__ void k_fill(float* __restrict__ p, float v, int n) {
    const int i = blockIdx.x * 256 + threadIdx.x;
    if (i < n) p[i] = v;
}

// BN batch stats: per-block register accumulation -> LDS -> global atomics
__global__ __launch_bounds__(256) void k_bn_stats(
    const float* __restrict__ y, float* __restrict__ bnsum,
    float* __restrict__ bnsq, int nRows, int rowsPerBlock) {
    __shared__ float ssum[128], ssq[128];
    const int tid = threadIdx.x;
    if (tid < 128) { ssum[tid] = 0.f; ssq[tid] = 0.f; }
    __syncthreads();
    const int c = tid & 127;
    const int half = tid >> 7;
    const int r0 = blockIdx.x * rowsPerBlock;
    int r1 = r0 + rowsPerBlock;
    if (r1 > nRows) r1 = nRows;
    float s = 0.f, q = 0.f;
    for (int r = r0 + half; r < r1; r += 2) {
        const float v = y[r * 128 + c];
        s += v;
        q += v * v;
    }
    atomicAdd(&ssum[c], s);
    atomicAdd(&ssq[c], q);
    __syncthreads();
    if (tid < 128) {
        atomicAdd(&bnsum[tid], ssum[tid]);
        atomicAdd(&bnsq[tid], ssq[tid]);
    }
}

__global__ void k_bn_final(const float* __restrict__ bnsum,
                           const float* __restrict__ bnsq,
                           const float* __restrict__ gamma,
                           const float* __restrict__ beta,
                           float* __restrict__ scale, float* __restrict__ shift,
                           float invN) {
    const int c = threadIdx.x;
    if (c < 128) {
        const float mu = bnsum[c] * invN;
        const float var = bnsq[c] * invN - mu * mu;
        const float rs = rsqrtf(var + 1e-5f) * gamma[c];
        scale[c] = rs;
        shift[c] = beta[c] - mu * rs;
    }
}

// normalize + relu + per-graph mean-pool; exploits sorted `batch` via
// run-length accumulation so atomics fire only on graph boundaries.
__global__ __launch_bounds__(256) void k_pool(
    const float* __restrict__ y, const int* __restrict__ batch,
    const float* __restrict__ scale, const float* __restrict__ shift,
    float* __restrict__ pool, float* __restrict__ cnt, int nRows,
    int rowsPerBlock) {
    const int tid = threadIdx.x;
    const int c = tid & 127;
    const int half = tid >> 7;
    const int r0 = blockIdx.x * rowsPerBlock;
    int r1 = r0 + rowsPerBlock;
    if (r1 > nRows) r1 = nRows;
    const float sc = scale[c], sh = shift[c];
    int curg = -1;
    float acc = 0.f, cacc = 0.f;
    for (int r = r0 + half; r < r1; r += 2) {
        const int g = batch[r];
        if (g != curg) {
            if (curg >= 0) {
                atomicAdd(&pool[curg * 128 + c], acc);
                if (c == 0) atomicAdd(&cnt[curg], cacc);
            }
            curg = g;
            acc = 0.f;
            cacc = 0.f;
        }
        const float v = y[r * 128 + c] * sc + sh;
        acc += fmaxf(v, 0.f);
        cacc += 1.f;
    }
    if (curg >= 0) {
        atomicAdd(&pool[curg * 128 + c], acc);
        if (c == 0) atomicAdd(&cnt[curg], cacc);
    }
}

// 200x10 fc2 + log_softmax: one thread per graph (trivial work)
__global__ void k_final(const float* __restrict__ pool,
                        const float* __restrict__ cnt,
                        const float* __restrict__ w2,
                        const float* __restrict__ b2, float* __restrict__ out) {
    const int g = blockIdx.x * 256 + threadIdx.x;
    if (g >= NUM_GRAPHS) return;
    const float inv = 1.f / fmaxf(cnt[g], 1.f);
    float lg[NUM_CLASSES];
    float mx = -1e30f;
    for (int cls = 0; cls < NUM_CLASSES; ++cls) {
        float s = b2[cls];
        for (int k = 0; k < 128; ++k)
            s += pool[g * 128 + k] * inv * w2[cls * 128 + k];
        lg[cls] = s;
        if (s > mx) mx = s;
    }
    float se = 0.f;
    for (int cls = 0; cls < NUM_CLASSES; ++cls) se += expf(lg[cls] - mx);
    const float lse = mx + logf(se);
    for (int cls = 0; cls < NUM_CLASSES; ++cls) out[g * 10 + cls] = lg[cls] - lse;
}

extern "C" void kernel_launch(void* const* d_in, const int* in_sizes, int n_in,
                              void* d_out, int out_size, void* d_ws,
                              size_t ws_size, hipStream_t stream) {
    const float* x = (const float*)d_in[0];
    const int* ei = (const int*)d_in[1];      // [2, E] int32
    const int* batch = (const int*)d_in[2];   // [N] int32, sorted
    const float* W = (const float*)d_in[3];   // [L,128,128]
    const float* wih = (const float*)d_in[4]; // [384,128] torch layout
    const float* whh = (const float*)d_in[5];
    const float* bih = (const float*)d_in[6];
    const float* bhh = (const float*)d_in[7];
    const float* fc1w = (const float*)d_in[8];  // [128,128] torch layout
    const float* fc1b = (const float*)d_in[9];
    const float* gamma = (const float*)d_in[10];
    const float* beta = (const float*)d_in[11];
    const float* fc2w = (const float*)d_in[12];
    const float* fc2b = (const float*)d_in[13];
    float* out = (float*)d_out;

    float* ws = (float*)d_ws;
    size_t o = 0;
    float* h = ws + o;     o += (size_t)N_NODES * 128;
    float* m = ws + o;     o += (size_t)N_NODES * 128;  // also fc1 output
    float* agg = ws + o;   o += (size_t)N_NODES * 128;
    float* Wp = ws + o;    o += (size_t)NUM_LAYERS * 128 * 128;  // packed GGNN
    float* wihP = ws + o;  o += 128 * 384;
    float* whhP = ws + o;  o += 128 * 384;
    float* fc1P = ws + o;  o += 128 * 128;
    float* bnsum = ws + o; o += 128;
    float* bnsq = ws + o;  o += 128;
    float* scale = ws + o; o += 128;
    float* shift = ws + o; o += 128;
    float* pool = ws + o;  o += NUM_GRAPHS * 128;
    float* cnt = ws + o;   o += NUM_GRAPHS;

    const int NH = N_NODES * 128;

    hipMemcpyAsync(h, x, sizeof(float) * NH, hipMemcpyDeviceToDevice, stream);

    // Pack all WMMA B operands into the interleaved K-pair feed layout.
    for (int l = 0; l < NUM_LAYERS; ++l)
        k_pack_direct<<<(128 * 128 + 255) / 256, 256, 0, stream>>>(
            W + (size_t)l * 128 * 128, Wp + (size_t)l * 128 * 128, 128);
    k_pack_torch<<<(384 * 128 + 255) / 256, 256, 0, stream>>>(wih, wihP, 384);
    k_pack_torch<<<(384 * 128 + 255) / 256, 256, 0, stream>>>(whh, whhP, 384);
    k_pack_torch<<<(128 * 128 + 255) / 256, 256, 0, stream>>>(fc1w, fc1P, 128);

    for (int l = 0; l < NUM_LAYERS; ++l) {
        k_gemm_nx128<<<N_NODES / 16, 256, 0, stream>>>(
            h, Wp + (size_t)l * 128 * 128, nullptr, m);
        k_fill<<<(NH + 255) / 256, 256, 0, stream>>>(agg, 0.f, NH);
        k_scatter<<<(N_EDGES * 32 + 255) / 256, 256, 0, stream>>>(m, ei, agg,
                                                                  N_EDGES);
        k_gru_fused<<<N_NODES / 16, 256, 0, stream>>>(agg, h, wihP, whhP, bih,
                                                      bhh);
    }

    // fc1 (+bias) into m
    k_gemm_nx128<<<N_NODES / 16, 256, 0, stream>>>(h, fc1P, fc1b, m);

    // batchnorm (training-mode batch stats)
    k_fill<<<1, 256, 0, stream>>>(bnsum, 0.f, 256);  // bnsum+bnsq contiguous
    k_bn_stats<<<625, 256, 0, stream>>>(m, bnsum, bnsq, N_NODES, 160);
    k_bn_final<<<1, 128, 0, stream>>>(bnsum, bnsq, gamma, beta, scale, shift,
                                      1.f / (float)N_NODES);

    // relu + mean pool
    k_fill<<<(NUM_GRAPHS * 128 + NUM_GRAPHS + 255) / 256, 256, 0, stream>>>(
        pool, 0.f, NUM_GRAPHS * 128 + NUM_GRAPHS);  // pool+cnt contiguous
    k_pool<<<400, 256, 0, stream>>>(m, batch, scale, shift, pool, cnt, N_NODES,
                                    250);

    // fc2 + log_softmax
    k_final<<<1, 256, 0, stream>>>(pool, cnt, fc2w, fc2b, out);
}